// ElectricPotential_51384988729986
// MI455X (gfx1250) — compile-verified
//
#include <hip/hip_runtime.h>
#include <hip/hip_bf16.h>
#include <math.h>

// ---------------------------------------------------------------------------
// ElectricPotential (e-place density + DCT spectral solve) for gfx1250.
// Heavy GEMMs (8 x 1024^3 fp32) run on V_WMMA_F32_16X16X4_F32 with 2x2
// register blocking per wave. Both A and B tiles are staged K-contiguous in
// LDS (B transposed, stride 34) so every WMMA fragment is one aligned
// ds_load_b64 — no repacking movs, conflict-free banks.
// ---------------------------------------------------------------------------

typedef __attribute__((ext_vector_type(2))) float v2f;
typedef __attribute__((ext_vector_type(8))) float v8f;

#define TN 1024               // M == N == 1024, square everything
#define NUM_MOVABLE 1000000
#define NUM_FILLER  300000
#define NUM_NODES   (NUM_MOVABLE + NUM_FILLER)
#define SQRT2_F     1.414213562f
#define TARGET_DENSITY 0.9f
#define PI_D 3.14159265358979323846

// ---------------------------------------------------------------------------
// DCT basis generation: C[u][m] = cos(pi*u*(2m+1)/(2L)), S = sin(...)
// Angles in double to match the float64 reference construction.
// ---------------------------------------------------------------------------
__global__ void gen_dct_kernel(float* __restrict__ Cmat, float* __restrict__ Smat) {
    int idx = blockIdx.x * blockDim.x + threadIdx.x;   // 1M threads
    int u = idx >> 10;
    int m = idx & 1023;
    double ang = PI_D * (double)u * (2.0 * (double)m + 1.0) / (2.0 * (double)TN);
    Cmat[idx] = (float)cos(ang);
    Smat[idx] = (float)sin(ang);
}

// ---------------------------------------------------------------------------
// Density scatter: each node adds overlap-weighted area into a 5x5 window.
// 4MB map is L2 resident; global_atomic_add_f32 path.
// ---------------------------------------------------------------------------
__global__ void scatter_density_kernel(const float* __restrict__ pos,
                                       const float* __restrict__ nsx,
                                       const float* __restrict__ nsy,
                                       float* __restrict__ dmap) {
    int t = blockIdx.x * blockDim.x + threadIdx.x;
    if (t >= NUM_NODES) return;
    float x  = pos[t];
    float y  = pos[NUM_NODES + t];
    float sx = nsx[t];
    float sy = nsy[t];
    float sxc = fmaxf(sx, SQRT2_F);
    float syc = fmaxf(sy, SQRT2_F);
    float ratio = (sx * sy) / (sxc * syc);
    if (t >= NUM_MOVABLE) ratio *= TARGET_DENSITY;

    int bx0 = (int)floorf(x);   // BSX = BSY = 1, XL = YL = 0
    int by0 = (int)floorf(y);

    int   ixs[5], iys[5];
    float oxs[5], oys[5];
#pragma unroll
    for (int i = 0; i < 5; ++i) {
        int ix = min(max(bx0 + i, 0), TN - 1);
        float lox = (float)ix;
        ixs[i] = ix;
        oxs[i] = fmaxf(fminf(x + sxc, lox + 1.0f) - fmaxf(x, lox), 0.0f);
        int iy = min(max(by0 + i, 0), TN - 1);
        float loy = (float)iy;
        iys[i] = iy;
        oys[i] = fmaxf(fminf(y + syc, loy + 1.0f) - fmaxf(y, loy), 0.0f);
    }
#pragma unroll
    for (int i = 0; i < 5; ++i) {
        float rx = ratio * oxs[i];
        int rowoff = ixs[i] * TN;
#pragma unroll
        for (int j = 0; j < 5; ++j) {
            atomicAdd(&dmap[rowoff + iys[j]], rx * oys[j]);
        }
    }
}

// ---------------------------------------------------------------------------
// Generic fp32 WMMA GEMM:  D = alpha * op(A) * op(B),  all matrices 1024x1024.
// Block tile: 64 x 64, Kt = 32. 128 threads = 4 waves in a 2x2 grid; each
// wave computes a 32x32 region as 2x2 accumulators of 16x16 (4 x v8f).
// As[64][34] row-major (rows x K); Bs[64][34] stores B TRANSPOSED (cols x K)
// so all fragments are contiguous 8B-aligned pairs -> single ds_load_b64.
// ---------------------------------------------------------------------------
__global__ __launch_bounds__(128) void gemm_wmma_f32_kernel(
    const float* __restrict__ A, const float* __restrict__ B,
    float* __restrict__ D, int transA, int transB, float alpha) {

    __shared__ float As[64][34];   // [row][k]  stride 34: aligned + conflict-free
    __shared__ float Bs[64][34];   // [col][k]  B stored K-contiguous (transposed)

    const int t    = threadIdx.x;
    const int lane = t & 31;
    const int wave = t >> 5;          // 0..3
    const int waveRow = wave >> 1;    // 0..1  (32-row slabs)
    const int waveCol = wave & 1;     // 0..1  (32-col slabs)
    const int blockRow0 = blockIdx.y * 64;
    const int blockCol0 = blockIdx.x * 64;

    const int mA   = lane & 15;            // A operand: lane carries M
    const int koff = (lane >> 4) * 2;      // lanes 0-15 -> K{0,1}, 16-31 -> K{2,3}
    const int am0  = waveRow * 32 + mA;          // A rows for acc row 0
    const int am1  = am0 + 16;                   // A rows for acc row 1
    const int bn0  = waveCol * 32 + (lane & 15); // B cols for acc col 0
    const int bn1  = bn0 + 16;                   // B cols for acc col 1

    v8f acc00 = {}, acc01 = {}, acc10 = {}, acc11 = {};

    for (int k0 = 0; k0 < TN; k0 += 32) {
        // ---- stage A tile (64 rows x 32 k), 16 elems/thread ----
        if (!transA) {
#pragma unroll
            for (int j = 0; j < 16; ++j) {
                int e = t + 128 * j;
                int i = e >> 5, k = e & 31;          // coalesced along k
                As[i][k] = A[(blockRow0 + i) * TN + (k0 + k)];
            }
        } else {
#pragma unroll
            for (int j = 0; j < 16; ++j) {
                int e = t + 128 * j;
                int i = e & 63, k = e >> 6;          // coalesced along i
                As[i][k] = A[(k0 + k) * TN + (blockRow0 + i)];
            }
        }
        // ---- stage B tile transposed: Bs[n][k], 16 elems/thread ----
        if (!transB) {
#pragma unroll
            for (int j = 0; j < 16; ++j) {
                int e = t + 128 * j;
                int k = e >> 6, n = e & 63;          // coalesced along n
                Bs[n][k] = B[(k0 + k) * TN + (blockCol0 + n)];
            }
        } else {
#pragma unroll
            for (int j = 0; j < 16; ++j) {
                int e = t + 128 * j;
                int k = e & 31, n = e >> 5;          // coalesced along k
                Bs[n][k] = B[(blockCol0 + n) * TN + (k0 + k)];
            }
        }
        __syncthreads();

        // ---- 8 K-steps x 4 WMMAs; each fragment = one aligned b64 load ----
#pragma unroll
        for (int kk = 0; kk < 32; kk += 4) {
            v2f a0 = *(const v2f*)&As[am0][kk + koff];
            v2f a1 = *(const v2f*)&As[am1][kk + koff];
            v2f b0 = *(const v2f*)&Bs[bn0][kk + koff];
            v2f b1 = *(const v2f*)&Bs[bn1][kk + koff];
            acc00 = __builtin_amdgcn_wmma_f32_16x16x4_f32(
                false, a0, false, b0, (short)0, acc00, false, false);
            acc01 = __builtin_amdgcn_wmma_f32_16x16x4_f32(
                false, a0, false, b1, (short)0, acc01, false, false);
            acc10 = __builtin_amdgcn_wmma_f32_16x16x4_f32(
                false, a1, false, b0, (short)0, acc10, false, false);
            acc11 = __builtin_amdgcn_wmma_f32_16x16x4_f32(
                false, a1, false, b1, (short)0, acc11, false, false);
        }
        __syncthreads();
    }

    // ---- epilogue: C/D layout -> lane = col, (lane>>4)*8 + r = row ----
    const int colBase0 = blockCol0 + waveCol * 32 + (lane & 15);
    const int rowBase0 = blockRow0 + waveRow * 32 + (lane >> 4) * 8;
#pragma unroll
    for (int r = 0; r < 8; ++r) {
        D[(rowBase0 + r) * TN + colBase0]             = alpha * acc00[r];
        D[(rowBase0 + r) * TN + colBase0 + 16]        = alpha * acc01[r];
        D[(rowBase0 + 16 + r) * TN + colBase0]        = alpha * acc10[r];
        D[(rowBase0 + 16 + r) * TN + colBase0 + 16]   = alpha * acc11[r];
    }
}

// ---------------------------------------------------------------------------
// Spectral scaling: from auv build the three RHS maps, folding in the
// row-0 / col-0 halving of auv.
//   X1 = a * wu * inv2,  X2 = a * wv * inv2,  X3 = a * inv2 * 2
// ---------------------------------------------------------------------------
__global__ void make_rhs_kernel(const float* __restrict__ auv,
                                float* __restrict__ X1,
                                float* __restrict__ X2,
                                float* __restrict__ X3) {
    int idx = blockIdx.x * blockDim.x + threadIdx.x;
    int u = idx >> 10;
    int v = idx & 1023;
    float a = auv[idx];
    if (u == 0) a *= 0.5f;
    if (v == 0) a *= 0.5f;
    const float w0 = (float)(2.0 * PI_D / (double)TN);
    float wu = (float)u * w0;
    float wv = (float)v * w0;
    float inv2 = (idx == 0) ? 0.0f : 2.0f / (wu * wu + wv * wv);
    X1[idx] = a * wu * inv2;
    X2[idx] = a * wv * inv2;
    X3[idx] = a * inv2 * 2.0f;
}

// ---------------------------------------------------------------------------
// energy = sum(potential .* dmap) -> d_out[0]
// ---------------------------------------------------------------------------
__global__ void energy_kernel(const float* __restrict__ pot,
                              const float* __restrict__ dmap,
                              float* __restrict__ out) {
    __shared__ float sdata[256];
    float s = 0.0f;
    for (int i = blockIdx.x * blockDim.x + threadIdx.x; i < TN * TN;
         i += gridDim.x * blockDim.x) {
        s += pot[i] * dmap[i];
    }
    sdata[threadIdx.x] = s;
    __syncthreads();
    for (int off = 128; off > 0; off >>= 1) {
        if (threadIdx.x < off) sdata[threadIdx.x] += sdata[threadIdx.x + off];
        __syncthreads();
    }
    if (threadIdx.x == 0) atomicAdd(out, sdata[0]);
}

// ---------------------------------------------------------------------------
// Orchestration.
//   inputs: pos (2*NUM_NODES f32), node_size_x, node_size_y, initial_density_map
//   output: [energy(1), field_map_x(1M), field_map_y(1M)]
//   workspace: 8 slots of 1024*1024 f32 = 32 MB
// ---------------------------------------------------------------------------
extern "C" void kernel_launch(void* const* d_in, const int* in_sizes, int n_in,
                              void* d_out, int out_size, void* d_ws, size_t ws_size,
                              hipStream_t stream) {
    const float* pos  = (const float*)d_in[0];
    const float* nsx  = (const float*)d_in[1];
    const float* nsy  = (const float*)d_in[2];
    const float* imap = (const float*)d_in[3];

    const size_t SZ = (size_t)TN * TN;
    float* ws   = (float*)d_ws;
    float* Cm   = ws;             // DCT cos basis (Cm == Cn, square)
    float* Sm   = ws + 1 * SZ;    // DCT sin basis
    float* dmap = ws + 2 * SZ;    // density map
    float* Tt   = ws + 3 * SZ;    // GEMM intermediate (left products)
    float* auv  = ws + 4 * SZ;    // spectral coefficients
    float* X1   = ws + 5 * SZ;
    float* X2   = ws + 6 * SZ;
    float* X3   = ws + 7 * SZ;
    float* pot  = auv;            // reuse: auv dead after make_rhs

    float* out = (float*)d_out;
    float* fx  = out + 1;
    float* fy  = out + 1 + SZ;

    // init: density map starts at initial_density_map; energy accumulator = 0
    hipMemcpyAsync(dmap, imap, SZ * sizeof(float), hipMemcpyDeviceToDevice, stream);
    hipMemsetAsync(out, 0, sizeof(float), stream);

    // DCT basis
    gen_dct_kernel<<<dim3((unsigned)(SZ / 256)), dim3(256), 0, stream>>>(Cm, Sm);

    // density scatter
    scatter_density_kernel<<<dim3((NUM_NODES + 255) / 256), dim3(256), 0, stream>>>(
        pos, nsx, nsy, dmap);

    const dim3 gg(TN / 64, TN / 64);   // (16, 16) blocks
    const dim3 gb(128);                // 4 waves
    const float auv_scale = 4.0f / ((float)TN * (float)TN);   // (2/M)*(2/N); BSX*BSY=1

    // forward 2D DCT: auv = scale * Cm @ dmap @ Cm^T
    gemm_wmma_f32_kernel<<<gg, gb, 0, stream>>>(Cm, dmap, Tt, 0, 0, 1.0f);
    gemm_wmma_f32_kernel<<<gg, gb, 0, stream>>>(Tt, Cm, auv, 0, 1, auv_scale);

    // spectral scalings (with row0/col0 halving folded in)
    make_rhs_kernel<<<dim3((unsigned)(SZ / 256)), dim3(256), 0, stream>>>(auv, X1, X2, X3);

    // field_map_x = Sm^T @ X1 @ Cm
    gemm_wmma_f32_kernel<<<gg, gb, 0, stream>>>(Sm, X1, Tt, 1, 0, 1.0f);
    gemm_wmma_f32_kernel<<<gg, gb, 0, stream>>>(Tt, Cm, fx, 0, 0, 1.0f);

    // field_map_y = Cm^T @ X2 @ Sm
    gemm_wmma_f32_kernel<<<gg, gb, 0, stream>>>(Cm, X2, Tt, 1, 0, 1.0f);
    gemm_wmma_f32_kernel<<<gg, gb, 0, stream>>>(Tt, Sm, fy, 0, 0, 1.0f);

    // potential = Cm^T @ X3 @ Cm ; energy = sum(potential .* dmap)
    gemm_wmma_f32_kernel<<<gg, gb, 0, stream>>>(Cm, X3, Tt, 1, 0, 1.0f);
    gemm_wmma_f32_kernel<<<gg, gb, 0, stream>>>(Tt, Cm, pot, 0, 0, 1.0f);
    energy_kernel<<<dim3(256), dim3(256), 0, stream>>>(pot, dmap, out);
}